// LMNet_62955630625293
// MI455X (gfx1250) — compile-verified
//
#include <hip/hip_runtime.h>

typedef __attribute__((ext_vector_type(2))) float v2f;
typedef __attribute__((ext_vector_type(8))) float v8f;

#define KK 32   // samples per event == wave32 lanes
#define FCH 16  // conv features

// ---------------- zero fill ----------------
__global__ __launch_bounds__(256) void zero_kernel(float* __restrict__ p, int n) {
    int i = blockIdx.x * blockDim.x + threadIdx.x;
    if (i < n) p[i] = 0.0f;
}

// ---------------- forward projection: ybar[e] = sum_k w[e,k]*x[idx[e,k]] + s[e]
// one wave per event, lane = k  (coalesced 128B idx + 128B w per wave).
// idx/w are single-use 256MB streams -> non-temporal loads (TH=NT) so the
// 192MB L2 stays dedicated to the hot x (8MB) gather target.
__global__ __launch_bounds__(256) void fwd_kernel(const float* __restrict__ x,
                                                  const int* __restrict__ idx,
                                                  const float* __restrict__ w,
                                                  const float* __restrict__ contam,
                                                  float* __restrict__ ybar, int E) {
    int lane = threadIdx.x & 31;
    int e = blockIdx.x * 8 + (threadIdx.x >> 5);
    if (e >= E) return;                       // wave-uniform
    int off = e * KK + lane;
    int ix = __builtin_nontemporal_load(idx + off);
    float wv = __builtin_nontemporal_load(w + off);
    float p = wv * x[ix];                     // x stays regular-temporal (L2 hot)
    p += __shfl_xor(p, 1);
    p += __shfl_xor(p, 2);
    p += __shfl_xor(p, 4);
    p += __shfl_xor(p, 8);
    p += __shfl_xor(p, 16);
    if (lane == 0) ybar[e] = p + contam[e];
}

// ---------------- back projection: bp[idx[e,k]] += w[e,k] / ybar[e]
__global__ __launch_bounds__(256) void bwd_kernel(const int* __restrict__ idx,
                                                  const float* __restrict__ w,
                                                  const float* __restrict__ ybar,
                                                  float* __restrict__ bp, int E) {
    int lane = threadIdx.x & 31;
    int e = blockIdx.x * 8 + (threadIdx.x >> 5);
    if (e >= E) return;
    int off = e * KK + lane;
    int ix = __builtin_nontemporal_load(idx + off);
    float wv = __builtin_nontemporal_load(w + off);
    float v = wv / ybar[e];                   // ybar broadcast (same addr all lanes)
    atomicAdd(&bp[ix], v);                    // L2-resident f32 atomic scatter
}

// ---------------- fused: x' = x - pre*(ao - bp);  4-layer 1x1x1 conv net; relu(x'-h)
// One wave handles a 16-voxel tile. Channel mixes (16x16) via V_WMMA_F32_16X16X4_F32.
// Layouts (wave32): n = lane&15 (voxel / N / A-row), half = lane>>4.
//   A 16x4 chunk kb : a[j]      = W[m=n][4kb + 2*half + j]
//   B 4x16 chunk kb : b[j]      = H[4kb + 2*half + j][n]
//   C/D 16x16       : acc[r]    = D[m = r + 8*half][n]
__global__ __launch_bounds__(256) void update_net_kernel(
    const float* x, float* __restrict__ bp,
    const float* __restrict__ ao, const float* __restrict__ pre,
    const float* __restrict__ w1, const float* __restrict__ b1,
    const float* __restrict__ w2, const float* __restrict__ b2,
    const float* __restrict__ pa, const float* __restrict__ w3,
    const float* __restrict__ b3, const float* __restrict__ w4,
    const float* __restrict__ b4, float* out,
    int iblk, int nTiles, int nvox) {
    int lane = threadIdx.x & 31;
    int half = lane >> 4;
    int n = lane & 15;
    int tile = blockIdx.x * 8 + (threadIdx.x >> 5);
    if (tile >= nTiles) return;               // wave-uniform: EXEC stays all-1s
    int vraw = tile * 16 + n;
    int v = vraw < nvox ? vraw : nvox - 1;    // clamp (nvox is 16-divisible here)
    bool live = (vraw < nvox) && (half == 0);

    // gradient step (both halves load the same 16 voxels -> no broadcast needed)
    float bpv = bp[v];
    float xnew = x[v] - pre[v] * (ao[v] - bpv);
    if (live) bp[v] = 0.0f;                   // reset accumulator for next block

    const float* W1 = w1 + iblk * FCH;
    const float* B1 = b1 + iblk * FCH;
    const float* W2 = w2 + iblk * FCH * FCH;  // [o][c] row-major
    const float* B2 = b2 + iblk * FCH;
    const float* W3 = w3 + iblk * FCH * FCH;
    const float* B3 = b3 + iblk * FCH;
    const float* W4 = w4 + iblk * FCH;
    float b4v = b4[iblk];
    float alpha = pa[iblk];

    // h1 directly in B layout: channel c = 4*(j>>1) + (j&1) + 2*half
    float bH[8];
#pragma unroll
    for (int j = 0; j < 8; ++j) {
        int c = 4 * (j >> 1) + (j & 1) + 2 * half;
        bH[j] = W1[c] * xnew + B1[c];
    }

    int m = n;  // A-matrix row (output channel index on lanes)
    // conv2: acc = W2 x H1 + b2, accumulate 4 chunks of K=4
    v8f acc;
#pragma unroll
    for (int r = 0; r < 8; ++r) acc[r] = B2[r + 8 * half];
#pragma unroll
    for (int kb = 0; kb < 4; ++kb) {
        v2f a; a[0] = W2[m * FCH + 4 * kb + 2 * half];
               a[1] = W2[m * FCH + 4 * kb + 2 * half + 1];
        v2f b; b[0] = bH[2 * kb]; b[1] = bH[2 * kb + 1];
        acc = __builtin_amdgcn_wmma_f32_16x16x4_f32(false, a, false, b,
                                                    (short)0, acc, false, false);
    }
    // PReLU on D layout
#pragma unroll
    for (int r = 0; r < 8; ++r) { float t = acc[r]; acc[r] = t >= 0.0f ? t : alpha * t; }

    // D layout -> B layout: exchange the 8 channel regs with partner half (lane^16)
    float ex[8];
#pragma unroll
    for (int r = 0; r < 8; ++r) ex[r] = __shfl_xor(acc[r], 16);
    float bH2[8];
#pragma unroll
    for (int j = 0; j < 8; ++j) {
        int c0 = 4 * (j >> 1) + (j & 1);      // target channel for half 0
        int c1 = c0 + 2;                      // target channel for half 1
        float v0 = (c0 < 8) ? acc[c0 & 7] : ex[c0 & 7];
        float v1 = (c1 >= 8) ? acc[c1 & 7] : ex[c1 & 7];
        bH2[j] = half ? v1 : v0;
    }

    // conv3: acc3 = W3 x H2 + b3
    v8f acc3;
#pragma unroll
    for (int r = 0; r < 8; ++r) acc3[r] = B3[r + 8 * half];
#pragma unroll
    for (int kb = 0; kb < 4; ++kb) {
        v2f a; a[0] = W3[m * FCH + 4 * kb + 2 * half];
               a[1] = W3[m * FCH + 4 * kb + 2 * half + 1];
        v2f b; b[0] = bH2[2 * kb]; b[1] = bH2[2 * kb + 1];
        acc3 = __builtin_amdgcn_wmma_f32_16x16x4_f32(false, a, false, b,
                                                     (short)0, acc3, false, false);
    }

    // conv4 (16 -> 1): per-lane partial over own 8 channels, then cross-half sum
    float pp = 0.0f;
#pragma unroll
    for (int r = 0; r < 8; ++r) pp += W4[r + 8 * half] * acc3[r];
    pp += __shfl_xor(pp, 16);
    float h4 = pp + b4v;
    float xo = xnew - h4;
    xo = xo > 0.0f ? xo : 0.0f;               // ReLU
    if (live) out[v] = xo;
}

extern "C" void kernel_launch(void* const* d_in, const int* in_sizes, int n_in,
                              void* d_out, int out_size, void* d_ws, size_t ws_size,
                              hipStream_t stream) {
    const float* x0     = (const float*)d_in[0];
    const int*   idx    = (const int*)d_in[1];
    const float* pw     = (const float*)d_in[2];
    const float* contam = (const float*)d_in[3];
    const float* ao     = (const float*)d_in[4];
    const float* pre    = (const float*)d_in[5];
    const float* w1     = (const float*)d_in[6];
    const float* b1     = (const float*)d_in[7];
    const float* w2     = (const float*)d_in[8];
    const float* b2     = (const float*)d_in[9];
    const float* pa     = (const float*)d_in[10];
    const float* w3     = (const float*)d_in[11];
    const float* b3     = (const float*)d_in[12];
    const float* w4     = (const float*)d_in[13];
    const float* b4     = (const float*)d_in[14];
    float* out = (float*)d_out;

    int nvox = in_sizes[0];
    int E = in_sizes[1] / KK;

    float* ybar = (float*)d_ws;       // E floats
    float* bp   = ybar + E;           // nvox floats

    int nTiles  = (nvox + 15) / 16;
    int eBlocks = (E + 7) / 8;        // 8 waves (events) per 256-thread block
    int tBlocks = (nTiles + 7) / 8;

    zero_kernel<<<(nvox + 255) / 256, 256, 0, stream>>>(bp, nvox);

    const float* xin = x0;
    for (int i = 0; i < 3; ++i) {
        fwd_kernel<<<eBlocks, 256, 0, stream>>>(xin, idx, pw, contam, ybar, E);
        bwd_kernel<<<eBlocks, 256, 0, stream>>>(idx, pw, ybar, bp, E);
        update_net_kernel<<<tBlocks, 256, 0, stream>>>(xin, bp, ao, pre,
            w1, b1, w2, b2, pa, w3, b3, w4, b4, out, i, nTiles, nvox);
        xin = out;  // d_out doubles as the x buffer for subsequent blocks
    }
}